// yolo_detect_target_22926535426313
// MI455X (gfx1250) — compile-verified
//
#include <hip/hip_runtime.h>
#include <stdint.h>

// ---------------- Problem constants (from reference) ----------------
#define CONF        0.25f
#define NCH         80              // channels per row
#define QPR         20              // float4 "quads" per row (80/4)
#define ROW_BYTES   320             // 80 * 4B, 16B aligned

// ---------------- Tiling ----------------
#define NTHREADS        256
#define TILE_ROWS       128         // rows staged per async tile
#define TILE_PITCH      336         // LDS row pitch: 84 dwords (pad 4) -> ~2-way bank conflicts
#define TILE_BYTES      (TILE_ROWS * TILE_PITCH)        // 43008
#define TILES_PER_CHUNK 4
#define CHUNK_ROWS      (TILE_ROWS * TILES_PER_CHUNK)   // 512 rows per block
#define IPT             ((TILE_ROWS * QPR) / NTHREADS)  // async issues per thread per tile = 10

#define USE_ASYNC 1

// gfx1250 async global->LDS copy, GVS mode: saddr(base SGPR pair) + vaddr(32b byte offset),
// vdst = per-lane LDS byte address (low 32 bits of the flat shared pointer).
__device__ __forceinline__ void async_g2l_b128(uint32_t lds_byte_addr, uint32_t g_off,
                                               const void* sbase) {
    asm volatile("global_load_async_to_lds_b128 %0, %1, %2"
                 :: "v"(lds_byte_addr), "v"(g_off), "s"(sbase)
                 : "memory");
}
__device__ __forceinline__ void wait_async(int more_in_flight) {
    // more_in_flight is block-uniform; exactly IPT loads per thread per non-empty tile.
    if (more_in_flight) asm volatile("s_wait_asynccnt 10" ::: "memory");
    else                asm volatile("s_wait_asynccnt 0"  ::: "memory");
}

extern "C" __global__ void __launch_bounds__(NTHREADS)
yolo_pass1(const float* __restrict__ post, const float* __restrict__ boxes,
           int nrows, float* __restrict__ ws_sum, int* __restrict__ ws_fail)
{
    __shared__ __align__(16) unsigned char tilebuf[2][TILE_BYTES];  // double buffer
    __shared__ float vrow[CHUNK_ROWS];
    __shared__ float redf[NTHREADS];
    __shared__ int   minFail;

    const int t = threadIdx.x;
    const int chunk = blockIdx.x;
    const int chunkStart = chunk * CHUNK_ROWS;
    const int rowsValid = min(CHUNK_ROWS, nrows - chunkStart);

    if (t == 0) minFail = CHUNK_ROWS;

    const uint32_t ldsBase = (uint32_t)(size_t)&tilebuf[0][0];

    // Issue one tile's async copies. Returns 1 if the tile is non-empty (issued
    // exactly IPT loads per thread), 0 otherwise. Row index is CLAMPED (not
    // skipped) in the tail so every thread always issues exactly IPT loads ->
    // per-wave ASYNCcnt is uniform and the partial waits below are exact.
    auto issueTile = [&](int tt) -> int {
        const int row0  = tt * TILE_ROWS;
        const int valid = min(TILE_ROWS, rowsValid - row0);
        if (valid <= 0) return 0;
        const char* gbase = (const char*)post + (size_t)(chunkStart + row0) * ROW_BYTES;
        const uint32_t bufBase = ldsBase + (uint32_t)((tt & 1) * TILE_BYTES);
#if USE_ASYNC
        #pragma unroll
        for (int k = 0; k < IPT; ++k) {
            const int q  = k * NTHREADS + t;         // flat quad index within tile
            const int r  = q / QPR;
            const int c  = q - r * QPR;
            const int rc = min(r, valid - 1);        // clamp: duplicate-copy of valid data
            const uint32_t loff = bufBase + (uint32_t)(rc * TILE_PITCH + c * 16);
            const uint32_t goff = (uint32_t)((rc * QPR + c) * 16);
            async_g2l_b128(loff, goff, gbase);
        }
#else
        #pragma unroll
        for (int k = 0; k < IPT; ++k) {
            const int q  = k * NTHREADS + t;
            const int r  = q / QPR;
            const int c  = q - r * QPR;
            const int rc = min(r, valid - 1);
            float4 f = *(const float4*)(gbase + (size_t)(rc * QPR + c) * 16);
            *(float4*)(tilebuf[tt & 1] + rc * TILE_PITCH + c * 16) = f;
        }
#endif
        return 1;
    };

    issueTile(0);
    __syncthreads();   // covers minFail init; async issue needs no barrier

    for (int tt = 0; tt < TILES_PER_CHUNK; ++tt) {
        // Prefetch next tile into the other buffer, then wait for current tile only.
        int nxt = 0;
        if (tt + 1 < TILES_PER_CHUNK) nxt = issueTile(tt + 1);
        wait_async(nxt);
        __syncthreads();      // all waves' copies for tile tt have landed

        // Process tile out of LDS: 2 lanes per row, 10 b128 reads each.
        const int tileRow0 = tt * TILE_ROWS;
        const int tileRowsValid = min(TILE_ROWS, rowsValid - tileRow0);
        const int r    = t >> 1;       // 0..127
        const int part = t & 1;
        float m = -__builtin_inff();
        if (r < tileRowsValid) {
            const float4* lrow = (const float4*)(tilebuf[tt & 1] + r * TILE_PITCH);
            #pragma unroll
            for (int j = 0; j < QPR / 2; ++j) {
                float4 f = lrow[part * (QPR / 2) + j];
                m = fmaxf(m, fmaxf(fmaxf(f.x, f.y), fmaxf(f.z, f.w)));
            }
        }
        m = fmaxf(m, __shfl_xor(m, 1));              // combine the two half-rows (wave32)

        if (part == 0 && r < tileRowsValid) {
            const int lr   = tileRow0 + r;           // chunk-local row
            const int grow = chunkStart + lr;
            float4 b = ((const float4*)boxes)[grow];
            vrow[lr] = m + (b.x + b.y) + (b.z + b.w);
            if (!(m >= CONF)) atomicMin(&minFail, lr);   // ds_min on LDS (NaN -> fail)
        }
        __syncthreads();      // reads of buf[tt&1] done before it is refilled at tt+2
    }

    // Chunk-local decoupled result: K = first local fail (or CHUNK_ROWS)
    const int K   = minFail;
    const int lim = min(K, rowsValid);
    float s = 0.0f;
    for (int i = t; i < lim; i += NTHREADS) s += vrow[i];

    redf[t] = s;
    __syncthreads();
    #pragma unroll
    for (int off = NTHREADS / 2; off > 0; off >>= 1) {
        if (t < off) redf[t] += redf[t + off];
        __syncthreads();
    }
    if (t == 0) {
        ws_sum[chunk]  = redf[0];
        ws_fail[chunk] = (K < rowsValid) ? (chunkStart + K) : 0x7FFFFFFF;
    }
}

extern "C" __global__ void __launch_bounds__(256)
yolo_pass2(const float* __restrict__ ws_sum, const int* __restrict__ ws_fail,
           int nChunks, float* __restrict__ out)
{
    __shared__ int   redi[256];
    __shared__ float redf[256];
    __shared__ int   s_cf;
    const int t = threadIdx.x;

    int mn = 0x7FFFFFFF;
    for (int i = t; i < nChunks; i += 256) mn = min(mn, ws_fail[i]);
    redi[t] = mn;
    __syncthreads();
    #pragma unroll
    for (int off = 128; off > 0; off >>= 1) {
        if (t < off) redi[t] = min(redi[t], redi[t + off]);
        __syncthreads();
    }
    if (t == 0) {
        const int Fr = redi[0];
        s_cf = (Fr == 0x7FFFFFFF) ? (nChunks - 1) : (Fr / CHUNK_ROWS);
    }
    __syncthreads();
    const int cf = s_cf;      // chunks 0..cf contribute (chunk cf stored its prefix-sum)

    float s = 0.0f;
    for (int i = t; i <= cf; i += 256) s += ws_sum[i];
    redf[t] = s;
    __syncthreads();
    #pragma unroll
    for (int off = 128; off > 0; off >>= 1) {
        if (t < off) redf[t] += redf[t + off];
        __syncthreads();
    }
    if (t == 0) out[0] = redf[0];
}

extern "C" void kernel_launch(void* const* d_in, const int* in_sizes, int n_in,
                              void* d_out, int out_size, void* d_ws, size_t ws_size,
                              hipStream_t stream)
{
    (void)n_in; (void)out_size; (void)ws_size;
    const float* post  = (const float*)d_in[0];
    const float* boxes = (const float*)d_in[1];
    const int nrows   = in_sizes[0] / NCH;
    const int nChunks = (nrows + CHUNK_ROWS - 1) / CHUNK_ROWS;

    float* ws_sum  = (float*)d_ws;
    int*   ws_fail = (int*)((char*)d_ws + (((size_t)nChunks * sizeof(float) + 255) & ~(size_t)255));

    yolo_pass1<<<nChunks, NTHREADS, 0, stream>>>(post, boxes, nrows, ws_sum, ws_fail);
    yolo_pass2<<<1, 256, 0, stream>>>(ws_sum, ws_fail, nChunks, (float*)d_out);
}